// QuantumAttention_65481071407658
// MI455X (gfx1250) — compile-verified
//
#include <hip/hip_runtime.h>

typedef __attribute__((ext_vector_type(2))) float v2f;
typedef __attribute__((ext_vector_type(8))) float v8f;

#define DEVFN static __device__ __forceinline__

// ---------------- RY on a k-register bit (wires 5..9) : pure VALU ----------------
template<int TB>
DEVFN void ry_k(float (&re)[32], float (&im)[32], float cw, float sw) {
#pragma unroll
  for (int k0 = 0; k0 < 32; ++k0) {
    if (k0 & (1 << TB)) continue;
    const int k1 = k0 | (1 << TB);
    float r0 = re[k0], r1 = re[k1];
    re[k0] = cw * r0 - sw * r1;
    re[k1] = sw * r0 + cw * r1;
    float i0 = im[k0], i1 = im[k1];
    im[k0] = cw * i0 - sw * i1;
    im[k1] = sw * i0 + cw * i1;
  }
}

// ------------- RY on a virtual lane bit (wires 0..4) : xor-shuffle butterfly -------------
DEVFN void ry_lane(float (&re)[32], float (&im)[32], float cw, float sw,
                   int lane, int vlane, int b, int mb) {
  const int src = lane ^ mb;                       // partner physical lane (L is GF(2)-linear)
  const float sgn = ((vlane >> b) & 1) ? sw : -sw; // a0: c*a0 - s*a1 ; a1: c*a1 + s*a0
#pragma unroll
  for (int k = 0; k < 32; ++k) {
    float pr = __shfl(re[k], src, 32);
    float pi = __shfl(im[k], src, 32);
    re[k] = cw * re[k] + sgn * pr;
    im[k] = cw * im[k] + sgn * pi;
  }
}

// ---------------- CNOT, control & target both k bits : free register swap ----------------
template<int CB, int TB>
DEVFN void cnot_kk(float (&re)[32], float (&im)[32]) {
#pragma unroll
  for (int k = 0; k < 32; ++k) {
    if (!(k & (1 << CB)) || (k & (1 << TB))) continue;
    const int k2 = k | (1 << TB);
    float t;
    t = re[k]; re[k] = re[k2]; re[k2] = t;
    t = im[k]; im[k] = im[k2]; im[k2] = t;
  }
}

// ---------- CNOT, control = virtual lane bit cb, target = k bit TB : predicated swap ----------
template<int TB>
DEVFN void cnot_lane_k(float (&re)[32], float (&im)[32], int vlane, int cb) {
  const bool c = ((vlane >> cb) & 1) != 0;
#pragma unroll
  for (int k = 0; k < 32; ++k) {
    if (k & (1 << TB)) continue;
    const int k2 = k | (1 << TB);
    float r0 = re[k], r1 = re[k2];
    re[k]  = c ? r1 : r0;
    re[k2] = c ? r0 : r1;
    float i0 = im[k], i1 = im[k2];
    im[k]  = c ? i1 : i0;
    im[k2] = c ? i0 : i1;
  }
}

// ---------- CNOT, control = k bit CB, target = virtual lane bit (phys mask mtb) ----------
template<int CB>
DEVFN void cnot_k_lane(float (&re)[32], float (&im)[32], int lane, int mtb) {
  const int src = lane ^ mtb;
#pragma unroll
  for (int k = 0; k < 32; ++k) {
    if (!(k & (1 << CB))) continue;
    re[k] = __shfl(re[k], src, 32);
    im[k] = __shfl(im[k], src, 32);
  }
}

// sign matrix entry for the WMMA B operand: columns 5..9 carry k-dependent Z signs,
// columns 0..4 and 10..15 are zero (lane-wire signs handled by shuffle reductions).
DEVFN float zsign(int k, int w) {
  if (w >= 5 && w < 10) return 1.0f - 2.0f * (float)((k >> (9 - w)) & 1);
  return 0.0f;
}

__global__ __launch_bounds__(256) void qsim_kernel(const float* __restrict__ x,
                                                   const float* __restrict__ wts,
                                                   float* __restrict__ out, int M) {
  const int lane = threadIdx.x & 31;
  const int wv   = threadIdx.x >> 5;
  const int m    = blockIdx.x * 8 + wv;   // one wave == one token; uniform per wave
  if (m >= M) return;                     // uniform exit -> surviving waves keep EXEC all-ones

  // ---- per-wire RX half-angle sin/cos (wave-uniform data) ----
  float ch[10], sh[10];
#pragma unroll
  for (int i = 0; i < 10; ++i) {
    float h = 0.5f * x[(size_t)m * 10 + i];
    __sincosf(h, &sh[i], &ch[i]);
  }

  // ---- build RX product state: amp(a) = mag(a) * (-i)^popcount(a) ----
  // amp index a = lane*32 + k : wires 0..4 <-> lane bits 4..0, wires 5..9 <-> k bits 4..0
  float lm = 1.0f;
#pragma unroll
  for (int j = 0; j < 5; ++j) lm *= ((lane >> (4 - j)) & 1) ? sh[j] : ch[j];
  const int lpc = __popc(lane);

  float re[32], im[32];
#pragma unroll
  for (int k = 0; k < 32; ++k) {
    float km = 1.0f;
#pragma unroll
    for (int j = 0; j < 5; ++j) km *= ((k >> (4 - j)) & 1) ? sh[5 + j] : ch[5 + j];
    const float mag = lm * km;
    const int ph = (lpc + __popc(k)) & 3;    // (-i)^ph
    re[k] = (ph == 0) ? mag : ((ph == 2) ? -mag : 0.0f);
    im[k] = (ph == 1) ? -mag : ((ph == 3) ? mag : 0.0f);
  }

  // virtual lane relabeling: vlane = L(lane); msk[b] = L^-1(e_b) (wave-uniform)
  int vlane = lane;
  int msk0 = 1, msk1 = 2, msk2 = 4, msk3 = 8, msk4 = 16;

  // =========================== layer 0 (CNOT range r=1) ===========================
  {
    float cw[10], sw[10];
#pragma unroll
    for (int i = 0; i < 10; ++i) __sincosf(0.5f * wts[i], &sw[i], &cw[i]);
    ry_lane(re, im, cw[0], sw[0], lane, vlane, 4, msk4);
    ry_lane(re, im, cw[1], sw[1], lane, vlane, 3, msk3);
    ry_lane(re, im, cw[2], sw[2], lane, vlane, 2, msk2);
    ry_lane(re, im, cw[3], sw[3], lane, vlane, 1, msk1);
    ry_lane(re, im, cw[4], sw[4], lane, vlane, 0, msk0);
    ry_k<4>(re, im, cw[5], sw[5]);
    ry_k<3>(re, im, cw[6], sw[6]);
    ry_k<2>(re, im, cw[7], sw[7]);
    ry_k<1>(re, im, cw[8], sw[8]);
    ry_k<0>(re, im, cw[9], sw[9]);
    // CNOT(0,1)(1,2)(2,3)(3,4): lane->lane, done algebraically (no data movement)
    vlane ^= ((vlane >> 4) & 1) << 3;  msk4 ^= msk3;
    vlane ^= ((vlane >> 3) & 1) << 2;  msk3 ^= msk2;
    vlane ^= ((vlane >> 2) & 1) << 1;  msk2 ^= msk1;
    vlane ^= ((vlane >> 1) & 1) << 0;  msk1 ^= msk0;
    cnot_lane_k<4>(re, im, vlane, 0);        // CNOT(4,5)
    cnot_kk<4, 3>(re, im);                   // CNOT(5,6)
    cnot_kk<3, 2>(re, im);                   // CNOT(6,7)
    cnot_kk<2, 1>(re, im);                   // CNOT(7,8)
    cnot_kk<1, 0>(re, im);                   // CNOT(8,9)
    cnot_k_lane<0>(re, im, lane, msk4);      // CNOT(9,0)
  }

  // =========================== layer 1 (CNOT range r=2) ===========================
  {
    float cw[10], sw[10];
#pragma unroll
    for (int i = 0; i < 10; ++i) __sincosf(0.5f * wts[10 + i], &sw[i], &cw[i]);
    ry_lane(re, im, cw[0], sw[0], lane, vlane, 4, msk4);
    ry_lane(re, im, cw[1], sw[1], lane, vlane, 3, msk3);
    ry_lane(re, im, cw[2], sw[2], lane, vlane, 2, msk2);
    ry_lane(re, im, cw[3], sw[3], lane, vlane, 1, msk1);
    ry_lane(re, im, cw[4], sw[4], lane, vlane, 0, msk0);
    ry_k<4>(re, im, cw[5], sw[5]);
    ry_k<3>(re, im, cw[6], sw[6]);
    ry_k<2>(re, im, cw[7], sw[7]);
    ry_k<1>(re, im, cw[8], sw[8]);
    ry_k<0>(re, im, cw[9], sw[9]);
    vlane ^= ((vlane >> 4) & 1) << 2;  msk4 ^= msk2;   // CNOT(0,2)
    vlane ^= ((vlane >> 3) & 1) << 1;  msk3 ^= msk1;   // CNOT(1,3)
    vlane ^= ((vlane >> 2) & 1) << 0;  msk2 ^= msk0;   // CNOT(2,4)
    cnot_lane_k<4>(re, im, vlane, 1);        // CNOT(3,5)
    cnot_lane_k<3>(re, im, vlane, 0);        // CNOT(4,6)
    cnot_kk<4, 2>(re, im);                   // CNOT(5,7)
    cnot_kk<3, 1>(re, im);                   // CNOT(6,8)
    cnot_kk<2, 0>(re, im);                   // CNOT(7,9)
    cnot_k_lane<1>(re, im, lane, msk4);      // CNOT(8,0)
    cnot_k_lane<0>(re, im, lane, msk3);      // CNOT(9,1)
  }

  // =========================== measurement ===========================
  float p[32];
#pragma unroll
  for (int k = 0; k < 32; ++k) p[k] = re[k] * re[k] + im[k] * im[k];

  const size_t obase = (size_t)m * 10;

  // ---- wires 0..4: sign depends only on (virtual) lane -> signed wave reductions ----
  float ptot = 0.0f;
#pragma unroll
  for (int k = 0; k < 32; ++k) ptot += p[k];
#pragma unroll
  for (int w = 0; w < 5; ++w) {
    const int b = 4 - w;
    float v = ((vlane >> b) & 1) ? -ptot : ptot;
#pragma unroll
    for (int s = 16; s > 0; s >>= 1) v += __shfl_xor(v, s, 32);
    if (lane == w) out[obase + w] = v;
  }

  // ---- wires 5..9: sign depends only on k -> per-wave GEMM on the matrix core ----
  // T(32x16) = P(32 lanes x 32 k) * Z(32 k x 16), then unsigned row-sum of T.
  // v_wmma_f32_16x16x4_f32: A 16x4 (lanes 0-15: M=lane, K=k0+{0,1}; lanes 16-31: K=k0+{2,3}),
  // B mirrors A with N across lanes; C/D 16x16 row-striped (8 VGPRs).
  const int wcol  = lane & 15;
  const int kbias = (lane & 16) ? 2 : 0;

  v2f B[8];
#pragma unroll
  for (int c = 0; c < 8; ++c) {
    const int k0 = 4 * c + kbias;
    B[c].x = zsign(k0, wcol);
    B[c].y = zsign(k0 + 1, wcol);
  }

  v8f acc0 = {0.f, 0.f, 0.f, 0.f, 0.f, 0.f, 0.f, 0.f};
  v8f acc1 = {0.f, 0.f, 0.f, 0.f, 0.f, 0.f, 0.f, 0.f};
  const int srcLo = lane & 15;   // hi half pulls from lane-16
  const int srcHi = lane | 16;   // lo half pulls from lane+16
#pragma unroll
  for (int c = 0; c < 8; ++c) {
    const int k0 = 4 * c;
    // M-tile 0: rows = lanes 0..15
    float t0 = __shfl(p[k0 + 2], srcLo, 32);
    float t1 = __shfl(p[k0 + 3], srcLo, 32);
    v2f a0;
    a0.x = (lane < 16) ? p[k0]     : t0;
    a0.y = (lane < 16) ? p[k0 + 1] : t1;
    acc0 = __builtin_amdgcn_wmma_f32_16x16x4_f32(false, a0, false, B[c],
                                                 (short)0, acc0, false, false);
    // M-tile 1: rows = lanes 16..31
    float u0 = __shfl(p[k0],     srcHi, 32);
    float u1 = __shfl(p[k0 + 1], srcHi, 32);
    v2f a1;
    a1.x = (lane < 16) ? u0 : p[k0 + 2];
    a1.y = (lane < 16) ? u1 : p[k0 + 3];
    acc1 = __builtin_amdgcn_wmma_f32_16x16x4_f32(false, a1, false, B[c],
                                                 (short)0, acc1, false, false);
  }

  // unsigned row-sum of both D tiles (lane signs are +1 for wires 5..9)
  float rs = 0.0f;
#pragma unroll
  for (int v = 0; v < 8; ++v) rs += acc0[v] + acc1[v];
  rs += __shfl_xor(rs, 16, 32);            // combine the two half-wave row groups
  if (lane >= 5 && lane < 10) out[obase + lane] = rs;   // column w = lane&15
}

extern "C" void kernel_launch(void* const* d_in, const int* in_sizes, int n_in,
                              void* d_out, int out_size, void* d_ws, size_t ws_size,
                              hipStream_t stream) {
  (void)n_in; (void)out_size; (void)d_ws; (void)ws_size;
  const float* x   = (const float*)d_in[0];   // (8,1024,10) f32
  const float* wts = (const float*)d_in[1];   // (2,10) f32
  float* out = (float*)d_out;                 // (8,1024,10) f32
  const int M = in_sizes[0] / 10;             // 8192 tokens
  const int blocks = (M + 7) / 8;             // 8 waves (tokens) per 256-thread block
  qsim_kernel<<<blocks, 256, 0, stream>>>(x, wts, out, M);
}